// GATBlock_63685775065348
// MI455X (gfx1250) — compile-verified
//
#include <hip/hip_runtime.h>

#define NN 50000
#define EE 800000
#define INC 256
#define HEADS 8
#define OUTC 32
#define HC 256            // HEADS*OUTC
#define NEG_SLOPE 0.2f
#define BN_EPS 1e-5f

typedef float v2f __attribute__((ext_vector_type(2)));
typedef float v8f __attribute__((ext_vector_type(8)));

// ---- float <-> order-preserving signed int (for atomicMax on floats) ----
__device__ __forceinline__ int f2ord(float f) {
    int i = __float_as_int(f);
    return i >= 0 ? i : (i ^ 0x7FFFFFFF);
}
__device__ __forceinline__ float ord2f(int i) {
    return __int_as_float(i >= 0 ? i : (i ^ 0x7FFFFFFF));
}

__device__ __forceinline__ void wait_asynccnt0() {
#if __has_builtin(__builtin_amdgcn_s_wait_asynccnt)
    __builtin_amdgcn_s_wait_asynccnt(0);
#else
    asm volatile("s_wait_asynccnt 0x0" ::: "memory");
#endif
}

// ---------------------------------------------------------------------------
// K0: zero/sentinel-init all accumulators (harness poisons ws/out with 0xAA)
// small = [ea_sum(1) | colsum(256) | colsumsq(256) | coef(8)] contiguous
// ---------------------------------------------------------------------------
__global__ void init_kernel(float* __restrict__ out, size_t n_out,
                            float* __restrict__ asum, int* __restrict__ amax, size_t nh,
                            float* __restrict__ small, size_t n_small) {
    size_t stride = (size_t)gridDim.x * blockDim.x;
    for (size_t i = (size_t)blockIdx.x * blockDim.x + threadIdx.x; i < n_out; i += stride) {
        out[i] = 0.0f;
        if (i < nh) { asum[i] = 0.0f; amax[i] = (int)0x80000000; }
        if (i < n_small) small[i] = 0.0f;
    }
}

// ---------------------------------------------------------------------------
// K1: sum(edge_attr) -> ea_sum[0]  (mean computed on the fly as sum/E)
// ---------------------------------------------------------------------------
__global__ void __launch_bounds__(256) ea_sum_kernel(const float* __restrict__ edge_attr,
                                                     float* __restrict__ ea_sum, int E) {
    __shared__ float red[256];
    float s = 0.0f;
    for (int i = blockIdx.x * 256 + threadIdx.x; i < E; i += gridDim.x * 256)
        s += edge_attr[i];
    red[threadIdx.x] = s;
    __syncthreads();
    for (int off = 128; off; off >>= 1) {
        if (threadIdx.x < off) red[threadIdx.x] += red[threadIdx.x + off];
        __syncthreads();
    }
    if (threadIdx.x == 0) atomicAdd(ea_sum, red[0]);
}

// ---------------------------------------------------------------------------
// K1b: coef[h] = sum_c W_edge[h*32+c] * att_edge[h*32+c]   (rank-1 he@att_edge)
// wave32: wave h == head h when blockDim == 256
// ---------------------------------------------------------------------------
__global__ void __launch_bounds__(256) coef_kernel(const float* __restrict__ W_edge,
                                                   const float* __restrict__ att_edge,
                                                   float* __restrict__ coef) {
    int t = threadIdx.x;
    float p = W_edge[t] * att_edge[t];
    for (int off = 16; off; off >>= 1) p += __shfl_down(p, off, 32);
    if ((t & 31) == 0) coef[t >> 5] = p;
}

// ---------------------------------------------------------------------------
// K2: h = x @ W  via V_WMMA_F32_16X16X4_F32 (fp32 GEMM, matches reference dtype)
// block = 256 threads = 8 waves; block tile = 16 rows x 128 cols (wave: 16x16)
// x tile staged in LDS with GLOBAL_LOAD_ASYNC_TO_LDS_B128 (ASYNCcnt path),
// LDS row stride 260 floats (1040 B, 16 B aligned) -> conflict-free A reads.
// Grid exactly tiles N (3125*16 == 50000), so no row guards needed on loads.
// ---------------------------------------------------------------------------
__global__ void __launch_bounds__(256) gemm_h_kernel(const float* __restrict__ x,
                                                     const float* __restrict__ W,
                                                     float* __restrict__ h, int N) {
    __shared__ float xs[16][260];
    const int tid  = threadIdx.x;
    const int wave = tid >> 5;
    const int lane = tid & 31;
    const int row0 = blockIdx.x * 16;
    const int col0 = (blockIdx.y * 8 + wave) * 16;

    // async-stage x tile: 16 rows x 256 cols = 1024 float4, 4 per thread,
    // global -> LDS directly (no VGPR round trip), tracked by ASYNCcnt
    for (int i = tid; i < 16 * 64; i += 256) {
        int r = i >> 6, c4 = i & 63;
        const float* gp = x + (size_t)(row0 + r) * INC + c4 * 4;
        unsigned lds_off = (unsigned)(size_t)(&xs[r][c4 * 4]);
        asm volatile("global_load_async_to_lds_b128 %0, %1, off"
                     :: "v"(lds_off), "v"(gp) : "memory");
    }
    wait_asynccnt0();
    __syncthreads();

    v8f acc = {0.f, 0.f, 0.f, 0.f, 0.f, 0.f, 0.f, 0.f};
    const int m     = lane & 15;         // A row (M)
    const int kbase = (lane >> 4) * 2;   // K pair: lanes 0-15 -> K{0,1}, 16-31 -> K{2,3}
    const int n     = lane & 15;         // B col (N)
    const float* Wcol = W + col0 + n;
    const float* xrow = &xs[m][0];

#pragma unroll 8
    for (int k0 = 0; k0 < INC; k0 += 4) {
        v2f a, b;
        a.x = xrow[k0 + kbase];
        a.y = xrow[k0 + kbase + 1];
        b.x = Wcol[(size_t)(k0 + kbase) * HC];
        b.y = Wcol[(size_t)(k0 + kbase + 1) * HC];
        // (neg_a, A, neg_b, B, c_mod, C, reuse_a, reuse_b)
        acc = __builtin_amdgcn_wmma_f32_16x16x4_f32(false, a, false, b,
                                                    (short)0, acc, false, false);
    }

    // C/D layout: VGPR r -> M = r + (lane<16 ? 0 : 8), N = lane&15
    const int sm = (lane < 16) ? 0 : 8;
#pragma unroll
    for (int r = 0; r < 8; ++r) {
        int gr = row0 + sm + r;
        if (gr < N) h[(size_t)gr * HC + col0 + n] = acc[r];
    }
}

// ---------------------------------------------------------------------------
// K3: a_src[n,h] = <h[n,h,:], att_src[h,:]>, same for a_dst.
// One block per node; wave h reduces head h via shuffles (wave32).
// ---------------------------------------------------------------------------
__global__ void __launch_bounds__(256) node_attn_kernel(const float* __restrict__ h,
                                                        const float* __restrict__ att_src,
                                                        const float* __restrict__ att_dst,
                                                        float* __restrict__ a_src,
                                                        float* __restrict__ a_dst, int N) {
    int nid = blockIdx.x;
    if (nid >= N) return;
    int t = threadIdx.x;
    float v  = h[(size_t)nid * HC + t];
    float ps = v * att_src[t];
    float pd = v * att_dst[t];
    for (int off = 16; off; off >>= 1) {
        ps += __shfl_down(ps, off, 32);
        pd += __shfl_down(pd, off, 32);
    }
    if ((t & 31) == 0) {
        int hd = t >> 5;
        a_src[(size_t)nid * HEADS + hd] = ps;
        a_dst[(size_t)nid * HEADS + hd] = pd;
    }
}

// ---------------------------------------------------------------------------
// K4: logits = leaky_relu(a_src[src]+a_dst[dst]+ea*coef); segment max via
// atomicMax on order-preserving ints. Edges e>=E are self loops (ea = mean).
// ---------------------------------------------------------------------------
__global__ void __launch_bounds__(256) edge_logits_kernel(const int* __restrict__ srcA,
                                                          const int* __restrict__ dstA,
                                                          const float* __restrict__ edge_attr,
                                                          const float* __restrict__ a_src,
                                                          const float* __restrict__ a_dst,
                                                          const float* __restrict__ coef,
                                                          const float* __restrict__ ea_sum,
                                                          float* __restrict__ logits,
                                                          int* __restrict__ amax,
                                                          int E, int N) {
    int e = blockIdx.x * blockDim.x + threadIdx.x;
    if (e >= E + N) return;
    int s, d; float ea;
    if (e < E) { s = srcA[e]; d = dstA[e]; ea = edge_attr[e]; }
    else       { s = d = e - E; ea = ea_sum[0] * (1.0f / (float)E); }
#pragma unroll
    for (int hd = 0; hd < HEADS; ++hd) {
        float t = a_src[(size_t)s * HEADS + hd] + a_dst[(size_t)d * HEADS + hd] + ea * coef[hd];
        t = t > 0.0f ? t : NEG_SLOPE * t;
        logits[(size_t)e * HEADS + hd] = t;
        atomicMax(&amax[(size_t)d * HEADS + hd], f2ord(t));
    }
}

// ---------------------------------------------------------------------------
// K5: alpha = exp(logit - amax[dst]); asum[dst] += alpha (in place over logits)
// ---------------------------------------------------------------------------
__global__ void __launch_bounds__(256) edge_exp_kernel(const int* __restrict__ dstA,
                                                       float* __restrict__ logits,
                                                       const int* __restrict__ amax,
                                                       float* __restrict__ asum,
                                                       int E, int N) {
    int e = blockIdx.x * blockDim.x + threadIdx.x;
    if (e >= E + N) return;
    int d = (e < E) ? dstA[e] : (e - E);
#pragma unroll
    for (int hd = 0; hd < HEADS; ++hd) {
        float mx = ord2f(amax[(size_t)d * HEADS + hd]);
        float ex = expf(logits[(size_t)e * HEADS + hd] - mx);
        logits[(size_t)e * HEADS + hd] = ex;
        atomicAdd(&asum[(size_t)d * HEADS + hd], ex);
    }
}

// ---------------------------------------------------------------------------
// K6: out[dst,:] += (alpha/asum[dst]) * h[src,:]  — one block per edge,
// thread c handles feature c (head = c>>5). Coalesced gather + coalesced
// global_atomic_add_f32; h and out are L2-resident (51 MB each << 192 MB).
// ---------------------------------------------------------------------------
__global__ void __launch_bounds__(256) edge_aggregate_kernel(const int* __restrict__ srcA,
                                                             const int* __restrict__ dstA,
                                                             const float* __restrict__ alpha,
                                                             const float* __restrict__ asum,
                                                             const float* __restrict__ h,
                                                             float* __restrict__ out,
                                                             int E, int N) {
    int e = blockIdx.x;
    int c = threadIdx.x;
    int s, d;
    if (e < E) { s = srcA[e]; d = dstA[e]; }
    else       { s = d = e - E; }
    int hd = c >> 5;
    float w = alpha[(size_t)e * HEADS + hd] / (asum[(size_t)d * HEADS + hd] + 1e-16f);
    atomicAdd(&out[(size_t)d * HC + c], w * h[(size_t)s * HC + c]);
}

// ---------------------------------------------------------------------------
// K7: per-column sum / sumsq over N rows (thread t == column t).
// Note: the reference adds a per-column constant bias before BN; that cancels
// exactly in (out - mean), so bias is omitted.
// ---------------------------------------------------------------------------
__global__ void __launch_bounds__(256) bn_reduce_kernel(const float* __restrict__ out,
                                                        float* __restrict__ colsum,
                                                        float* __restrict__ colsumsq, int N) {
    int t = threadIdx.x;
    float s = 0.0f, s2 = 0.0f;
    for (int n = blockIdx.x; n < N; n += gridDim.x) {
        float v = out[(size_t)n * HC + t];
        s += v;
        s2 += v * v;
    }
    atomicAdd(&colsum[t], s);
    atomicAdd(&colsumsq[t], s2);
}

// ---------------------------------------------------------------------------
// K8: BN (batch stats) + ELU, in place.
// ---------------------------------------------------------------------------
__global__ void __launch_bounds__(256) bn_elu_kernel(float* __restrict__ out,
                                                     const float* __restrict__ colsum,
                                                     const float* __restrict__ colsumsq,
                                                     const float* __restrict__ gamma,
                                                     const float* __restrict__ beta, int N) {
    size_t i = (size_t)blockIdx.x * blockDim.x + threadIdx.x;
    if (i >= (size_t)N * HC) return;
    int c = (int)(i & (HC - 1));
    float invN = 1.0f / (float)N;
    float mean = colsum[c] * invN;
    float var  = colsumsq[c] * invN - mean * mean;
    float v = (out[i] - mean) * rsqrtf(var + BN_EPS) * gamma[c] + beta[c];
    out[i] = v > 0.0f ? v : expm1f(v);
}

// ---------------------------------------------------------------------------
extern "C" void kernel_launch(void* const* d_in, const int* in_sizes, int n_in,
                              void* d_out, int out_size, void* d_ws, size_t ws_size,
                              hipStream_t stream) {
    const float* x          = (const float*)d_in[0];
    const int*   edge_index = (const int*)  d_in[1];   // [2,E] : src=ptr, dst=ptr+E
    const float* edge_attr  = (const float*)d_in[2];
    const float* W          = (const float*)d_in[3];
    const float* att_src    = (const float*)d_in[4];
    const float* att_dst    = (const float*)d_in[5];
    const float* W_edge     = (const float*)d_in[6];
    const float* att_edge   = (const float*)d_in[7];
    // d_in[8] = bias  : cancels exactly in BatchNorm -> unused
    const float* gamma      = (const float*)d_in[9];
    const float* beta       = (const float*)d_in[10];
    float* out = (float*)d_out;

    const int*   srcA = edge_index;
    const int*   dstA = edge_index + EE;

    // ---- workspace layout (floats) ----
    float* ws     = (float*)d_ws;
    float* h      = ws;                                  // N*256
    float* a_src  = h      + (size_t)NN * HC;            // N*8
    float* a_dst  = a_src  + (size_t)NN * HEADS;         // N*8
    int*   amax   = (int*)(a_dst + (size_t)NN * HEADS);  // N*8
    float* asum   = (float*)(amax + (size_t)NN * HEADS); // N*8
    float* logits = asum   + (size_t)NN * HEADS;         // (E+N)*8
    float* smallb = logits + (size_t)(EE + NN) * HEADS;  // 521 floats
    float* ea_sum   = smallb;
    float* colsum   = smallb + 1;
    float* colsumsq = smallb + 257;
    float* coef     = smallb + 513;

    // K0: init accumulators
    init_kernel<<<4096, 256, 0, stream>>>(out, (size_t)NN * HC,
                                          asum, amax, (size_t)NN * HEADS,
                                          smallb, (size_t)521);
    // K1: edge_attr mean (as sum), K1b: rank-1 edge coefficients
    ea_sum_kernel<<<1024, 256, 0, stream>>>(edge_attr, ea_sum, EE);
    coef_kernel<<<1, 256, 0, stream>>>(W_edge, att_edge, coef);

    // K2: h = x @ W (WMMA fp32, async-to-LDS staging)
    dim3 ggrid((NN + 15) / 16, 2);
    gemm_h_kernel<<<ggrid, 256, 0, stream>>>(x, W, h, NN);

    // K3: per-node attention projections
    node_attn_kernel<<<NN, 256, 0, stream>>>(h, att_src, att_dst, a_src, a_dst, NN);

    // K4-K6: segment softmax + weighted aggregation over E+N edges
    int nE = EE + NN;
    int eblk = (nE + 255) / 256;
    edge_logits_kernel<<<eblk, 256, 0, stream>>>(srcA, dstA, edge_attr, a_src, a_dst,
                                                 coef, ea_sum, logits, amax, EE, NN);
    edge_exp_kernel<<<eblk, 256, 0, stream>>>(dstA, logits, amax, asum, EE, NN);
    edge_aggregate_kernel<<<nE, 256, 0, stream>>>(srcA, dstA, logits, asum, h, out, EE, NN);

    // K7-K8: BatchNorm (batch stats) + ELU
    bn_reduce_kernel<<<512, 256, 0, stream>>>(out, colsum, colsumsq, NN);
    bn_elu_kernel<<<(NN * HC + 255) / 256, 256, 0, stream>>>(out, colsum, colsumsq,
                                                             gamma, beta, NN);
}